// MMFS_74801150427609
// MI455X (gfx1250) — compile-verified
//
#include <hip/hip_runtime.h>
#include <cstddef>

// ---------------- problem constants (from reference) ----------------
#define NB    2
#define LQ    576
#define NIMG  8
#define DIM   256
#define NH    8
#define NP    8
#define NLV   4            // NL
#define DH    32           // DIM / NH
#define NLEV  32           // NIMG * NLV
#define HWTOT 1360         // 32^2+16^2+8^2+4^2
#define VROWS (NIMG*HWTOT) // 10880 value rows per batch
#define AWC   288          // NH*NLV*(NP+1)
#define SOC   128          // NH*NP*2
#define NEG_LOG_NLEV (-3.4657359027997265f)  // -log(32)

typedef __attribute__((ext_vector_type(2))) float v2f;
typedef __attribute__((ext_vector_type(8))) float v8f;

// ---------------------------------------------------------------
// fp32 WMMA GEMM:  C[M,N] = A[M,K] @ W[K,N] + bias[N]
// Each wave computes a 16 x (16*NT) strip of C with NT independent
// accumulators: the A fragment is loaded once per k-step and reused
// across NT wmmas (fewer loads/wmma, independent dep chains for the
// XDL pipe). Requires M%16==0, N%(16*NT)==0, K%4==0.
//
// A frag (16x4 f32, ISA layout): lanes 0-15 -> M=lane, regs {K=0,K=1};
//                                lanes 16-31 -> M=lane-16, regs {K=2,K=3}.
// B frag (4x16 f32, mirrored):   lanes 0-15 -> N=lane, regs {K=0,K=1};
//                                lanes 16-31 -> N=lane-16, regs {K=2,K=3}.
// C/D (16x16 f32): reg r holds row M = r + 8*(lane>=16), col N = lane&15.
// ---------------------------------------------------------------
template <int NT>
__global__ void gemm_f32_wmma(const float* __restrict__ A,
                              const float* __restrict__ W,
                              const float* __restrict__ bias,
                              float* __restrict__ C,
                              int M, int N, int K)
{
    const int lane  = threadIdx.x & 31;
    const int wave  = threadIdx.x >> 5;
    const int wpb   = blockDim.x >> 5;
    const int tilesN = N / (16 * NT);
    const int tileId = blockIdx.x * wpb + wave;
    if (tileId >= (M >> 4) * tilesN) return;

    const int m0 = (tileId / tilesN) << 4;
    const int n0 = (tileId % tilesN) * (16 * NT);

    const int halfSel = lane >> 4;            // 0 or 1
    const int am  = m0 + (lane & 15);
    const int k01 = halfSel << 1;             // 0 or 2

    const float* __restrict__ Arow = A + (size_t)am * K + k01;
    const float* __restrict__ Bcol = W + (size_t)k01 * N + n0 + (lane & 15);

    v8f acc[NT];
#pragma unroll
    for (int t = 0; t < NT; ++t) acc[t] = v8f{};

#pragma unroll 2
    for (int k = 0; k < K; k += 4) {
        v2f a;
        a.x = Arow[k + 0];
        a.y = Arow[k + 1];
        v2f b[NT];
#pragma unroll
        for (int t = 0; t < NT; ++t) {
            b[t].x = Bcol[(size_t)k * N + 16 * t];
            b[t].y = Bcol[(size_t)(k + 1) * N + 16 * t];
        }
#pragma unroll
        for (int t = 0; t < NT; ++t) {
            acc[t] = __builtin_amdgcn_wmma_f32_16x16x4_f32(
                /*neg_a*/false, a, /*neg_b*/false, b[t],
                /*c_mod*/(short)0, acc[t], /*reuse_a*/false, /*reuse_b*/false);
        }
    }

    const int cm0 = m0 + (halfSel << 3);
#pragma unroll
    for (int t = 0; t < NT; ++t) {
        const int cn = n0 + 16 * t + (lane & 15);
        const float bb = bias ? bias[cn] : 0.0f;
#pragma unroll
        for (int r = 0; r < 8; ++r)
            C[(size_t)(cm0 + r) * N + cn] = acc[t][r] + bb;
    }
}

// ---------------------------------------------------------------
// q[b,img,l,d] = qd[b,l,d] + relpos_emb[relpos(b,img), d]
// relpos = (tot + 1 - prefix_incl) * am
// ---------------------------------------------------------------
__global__ void build_q_kernel(const float* __restrict__ qd,
                               const float* __restrict__ relpos_emb,
                               const int*   __restrict__ am,
                               float* __restrict__ q)
{
    int idx = blockIdx.x * blockDim.x + threadIdx.x;
    if (idx >= NB * NIMG * LQ * DIM) return;
    int d = idx & (DIM - 1);
    int t = idx >> 8;               // DIM==256
    int l = t % LQ;  t /= LQ;
    int img = t % NIMG;
    int b   = t / NIMG;

    int tot = 0, prev = 0;
#pragma unroll
    for (int i = 0; i < NIMG; ++i) {
        int a = am[b * NIMG + i];
        tot += a;
        if (i <= img) prev += a;
    }
    int rp = (tot + 1 - prev) * am[b * NIMG + img];
    q[idx] = qd[((size_t)(b * LQ + l)) * DIM + d] + relpos_emb[(size_t)rp * DIM + d];
}

// ---------------------------------------------------------------
// Softmax over the 288 = NLEV*(NP+1) logits per (b,l,h) row.
// One wave per row; lane <-> (img = lane>>2, lvl = lane&3), 9 entries/lane.
// Writes normalized aw_s[(b,l,h), lev, j] and the ignore-weight sum.
// ---------------------------------------------------------------
__global__ void softmax_kernel(const float* __restrict__ aw_raw,
                               const int*   __restrict__ amask,
                               float* __restrict__ aw_s,
                               float* __restrict__ aw_ig)
{
    int gwave = (blockIdx.x * blockDim.x + threadIdx.x) >> 5;
    int lane  = threadIdx.x & 31;
    if (gwave >= NB * LQ * NH) return;
    int h = gwave & (NH - 1);
    int t = gwave >> 3;
    int l = t % LQ;
    int b = t / LQ;
    int img = lane >> 2;
    int lvl = lane & 3;

    const float* src = aw_raw
        + ((size_t)((b * NIMG + img) * LQ + l)) * AWC
        + h * (NLV * (NP + 1)) + lvl * (NP + 1);
    float bias = amask[b * NIMG + img] ? 0.0f : -10000.0f;

    float v[NP + 1];
#pragma unroll
    for (int j = 0; j < NP; ++j) v[j] = src[j] + bias;
    v[NP] = NEG_LOG_NLEV;          // exact override, bias not applied

    float m = v[0];
#pragma unroll
    for (int j = 1; j <= NP; ++j) m = fmaxf(m, v[j]);
    for (int off = 16; off; off >>= 1) m = fmaxf(m, __shfl_xor(m, off, 32));

    float s = 0.0f;
#pragma unroll
    for (int j = 0; j <= NP; ++j) { v[j] = __expf(v[j] - m); s += v[j]; }
    for (int off = 16; off; off >>= 1) s += __shfl_xor(s, off, 32);
    float inv = 1.0f / s;

    float* dst = aw_s + (size_t)gwave * AWC + lane * (NP + 1);
#pragma unroll
    for (int j = 0; j <= NP; ++j) dst[j] = v[j] * inv;

    float wig = v[NP] * inv;
    for (int off = 16; off; off >>= 1) wig += __shfl_xor(wig, off, 32);
    if (lane == 0) aw_ig[gwave] = wig;
}

// ---------------------------------------------------------------
// Deformable sampling: one wave per (b,l,h); lane = channel dh (DH==32).
// value buffer (22 MB) is L2-resident; corner reads are 128B coalesced.
// ---------------------------------------------------------------
__global__ void sample_kernel(const float* __restrict__ value,
                              const float* __restrict__ aw_s,
                              const float* __restrict__ aw_ig,
                              const float* __restrict__ so_raw,
                              const float* __restrict__ refp,
                              const int*   __restrict__ ss,
                              const int*   __restrict__ lsi,
                              const float* __restrict__ scale_ratios,
                              const float* __restrict__ ignore_token,
                              float* __restrict__ attn_out)
{
    int gwave = (blockIdx.x * blockDim.x + threadIdx.x) >> 5;
    int lane  = threadIdx.x & 31;       // dh
    if (gwave >= NB * LQ * NH) return;
    int h = gwave & (NH - 1);
    int t = gwave >> 3;
    int l = t % LQ;
    int b = t / LQ;

    const float* awr = aw_s + (size_t)gwave * AWC;
    const float* rp  = refp + ((size_t)(b * LQ + l)) * NLEV * 2;

    float acc = 0.0f;
    for (int lev = 0; lev < NLEV; ++lev) {
        const int img = lev >> 2;
        const int lvl = lev & 3;
        const int Hl = ss[lev * 2 + 0];
        const int Wl = ss[lev * 2 + 1];
        const int s0 = lsi[lev];
        const float sr = scale_ratios[lvl];
        const float rx = rp[lev * 2 + 0];
        const float ry = rp[lev * 2 + 1];
        const float* so = so_raw
            + ((size_t)((b * NIMG + img) * LQ + l)) * SOC + h * (NP * 2);
        const float* vbase = value
            + ((size_t)b * VROWS + s0) * DIM + h * DH + lane;

        for (int p = 0; p < NP; ++p) {
            float w  = awr[lev * (NP + 1) + p];
            float sx = so[p * 2 + 0] * sr;
            float sy = so[p * 2 + 1] * sr;
            float x = (rx + sx / (float)Wl) * (float)Wl - 0.5f;
            float y = (ry + sy / (float)Hl) * (float)Hl - 0.5f;
            float xf = floorf(x), yf = floorf(y);
            float wx = x - xf,    wy = y - yf;
            int x0 = (int)xf, y0 = (int)yf;

            auto corner = [&](int xi, int yi, float cw) -> float {
                bool valid = (xi >= 0) & (xi < Wl) & (yi >= 0) & (yi < Hl);
                int xc = min(max(xi, 0), Wl - 1);
                int yc = min(max(yi, 0), Hl - 1);
                float smp = vbase[(size_t)(yc * Wl + xc) * DIM];
                return valid ? cw * smp : 0.0f;
            };
            float sum = corner(x0,     y0,     (1.0f - wx) * (1.0f - wy))
                      + corner(x0 + 1, y0,     wx          * (1.0f - wy))
                      + corner(x0,     y0 + 1, (1.0f - wx) * wy)
                      + corner(x0 + 1, y0 + 1, wx          * wy);
            acc += w * sum;
        }
    }
    acc += aw_ig[gwave] * ignore_token[h * DH + lane];
    attn_out[((size_t)(b * LQ + l)) * DIM + h * DH + lane] = acc;
}

// ---------------------------------------------------------------
// launch
// ---------------------------------------------------------------
extern "C" void kernel_launch(void* const* d_in, const int* in_sizes, int n_in,
                              void* d_out, int out_size, void* d_ws, size_t ws_size,
                              hipStream_t stream)
{
    (void)in_sizes; (void)n_in; (void)out_size; (void)ws_size;

    const float* query   = (const float*)d_in[0];
    const float* refp    = (const float*)d_in[1];
    const float* in_flat = (const float*)d_in[2];
    const int*   ss      = (const int*)  d_in[3];
    const int*   lsi     = (const int*)  d_in[4];
    const int*   amask   = (const int*)  d_in[5];
    const float* Wso     = (const float*)d_in[6];
    const float* bso     = (const float*)d_in[7];
    const float* Wdom    = (const float*)d_in[8];
    const float* bdom    = (const float*)d_in[9];
    const float* Wa      = (const float*)d_in[10];
    const float* ba      = (const float*)d_in[11];
    const float* Wv      = (const float*)d_in[12];
    const float* bv      = (const float*)d_in[13];
    const float* Wo      = (const float*)d_in[14];
    const float* bo      = (const float*)d_in[15];
    const float* relpos  = (const float*)d_in[16];
    const float* ign     = (const float*)d_in[17];
    const float* sratio  = (const float*)d_in[18];
    float* out = (float*)d_out;

    // workspace layout (floats)
    float* ws = (float*)d_ws;
    float* w_value = ws;                                    // 21760*256
    float* w_qd    = w_value + (size_t)NB*VROWS*DIM;        // 1152*256
    float* w_q     = w_qd    + (size_t)NB*LQ*DIM;           // 9216*256
    float* w_so    = w_q     + (size_t)NB*NIMG*LQ*DIM;      // 9216*128
    float* w_awr   = w_so    + (size_t)NB*NIMG*LQ*SOC;      // 9216*288
    float* w_aws   = w_awr   + (size_t)NB*NIMG*LQ*AWC;      // 9216*288
    float* w_awig  = w_aws   + (size_t)NB*LQ*NH*AWC;        // 9216
    float* w_attn  = w_awig  + (size_t)NB*LQ*NH;            // 1152*256

    const int THREADS = 256;               // 8 waves / block
    auto gemmBlocks = [](int M, int N, int nt) {
        return ((M >> 4) * (N / (16 * nt)) + 7) / 8;
    };

    // 1) value = input_flatten @ Wv + bv        (M=21760, N=256, NT=4)
    gemm_f32_wmma<4><<<gemmBlocks(NB*VROWS, DIM, 4), THREADS, 0, stream>>>(
        in_flat, Wv, bv, w_value, NB*VROWS, DIM, DIM);

    // 2) qd = query @ Wdom + bdom               (M=1152, N=256, NT=4)
    gemm_f32_wmma<4><<<gemmBlocks(NB*LQ, DIM, 4), THREADS, 0, stream>>>(
        query, Wdom, bdom, w_qd, NB*LQ, DIM, DIM);

    // 3) q = qd + relpos_emb[relpos]
    {
        int total = NB * NIMG * LQ * DIM;
        build_q_kernel<<<(total + THREADS - 1) / THREADS, THREADS, 0, stream>>>(
            w_qd, relpos, amask, w_q);
    }

    // 4) so = q @ Wso + bso                      (M=9216, N=128, NT=4)
    gemm_f32_wmma<4><<<gemmBlocks(NB*NIMG*LQ, SOC, 4), THREADS, 0, stream>>>(
        w_q, Wso, bso, w_so, NB*NIMG*LQ, SOC, DIM);

    // 5) aw = q @ Wa + ba                        (M=9216, N=288=9*32, NT=2)
    gemm_f32_wmma<2><<<gemmBlocks(NB*NIMG*LQ, AWC, 2), THREADS, 0, stream>>>(
        w_q, Wa, ba, w_awr, NB*NIMG*LQ, AWC, DIM);

    // 6) masked softmax with ignore slot
    {
        int waves = NB * LQ * NH;                       // 9216
        softmax_kernel<<<(waves * 32 + THREADS - 1) / THREADS, THREADS, 0, stream>>>(
            w_awr, amask, w_aws, w_awig);
    }

    // 7) deformable bilinear sampling + ignore token
    {
        int waves = NB * LQ * NH;                       // 9216
        sample_kernel<<<(waves * 32 + THREADS - 1) / THREADS, THREADS, 0, stream>>>(
            w_value, w_aws, w_awig, w_so, refp, ss, lsi, sratio, ign, w_attn);
    }

    // 8) out = attn @ Wo + bo                    (M=1152, N=256, NT=4)
    gemm_f32_wmma<4><<<gemmBlocks(NB*LQ, DIM, 4), THREADS, 0, stream>>>(
        w_attn, Wo, bo, out, NB*LQ, DIM, DIM);
}